// TwoLayerGCNLinearHead_19782619365932
// MI455X (gfx1250) — compile-verified
//
#include <hip/hip_runtime.h>
#include <hip/hip_bf16.h>

#define N_NODES  100000
#define N_EDGES  1600000
#define F_DIM    128
#define N_GRAPHS 64
#define N_LABELS 20

typedef __attribute__((ext_vector_type(2))) float v2f;
typedef __attribute__((ext_vector_type(8))) float v8f;

__device__ __forceinline__ void atom_add_f32(float* p, float v) {
    __hip_atomic_fetch_add(p, v, __ATOMIC_RELAXED, __HIP_MEMORY_SCOPE_AGENT);
}

// async global->LDS copy of 16B, tracked by ASYNCcnt (gfx1250)
__device__ __forceinline__ void async_copy_b128(unsigned lds_off, const void* gptr) {
    asm volatile("global_load_async_to_lds_b128 %0, %1, off"
                 :: "v"(lds_off), "v"((unsigned long long)(uintptr_t)gptr)
                 : "memory");
}
__device__ __forceinline__ void wait_async0() {
    asm volatile("s_wait_asynccnt 0x0" ::: "memory");
}

// ---------------------------------------------------------------------------
// 1) h[i,:] = emb[x[i],:]   (32 lanes per node, float4 each)
// ---------------------------------------------------------------------------
__global__ __launch_bounds__(256)
void gather_embed_kernel(const int* __restrict__ x, const float* __restrict__ emb,
                         float* __restrict__ H) {
    int tid = blockIdx.x * 256 + threadIdx.x;
    if (tid >= N_NODES * 32) return;
    int i = tid >> 5;
    int c = (tid & 31) << 2;
    int xi = x[i];
    float4 v = *(const float4*)(emb + (long)xi * F_DIM + c);
    *(float4*)(H + (long)i * F_DIM + c) = v;
}

// ---------------------------------------------------------------------------
// 2) agg[dst,:] += h[src,:]  (32 lanes per edge, float4 each -> 4 fp32 atomics)
// ---------------------------------------------------------------------------
__global__ __launch_bounds__(256)
void scatter_edges_kernel(const float* __restrict__ H, const int* __restrict__ src,
                          const int* __restrict__ dst, float* __restrict__ Agg) {
    long tid = (long)blockIdx.x * 256 + threadIdx.x;
    if (tid >= (long)N_EDGES * 32) return;
    int e = (int)(tid >> 5);
    int c = ((int)tid & 31) << 2;
    int s = src[e], d = dst[e];
    float4 v = *(const float4*)(H + (long)s * F_DIM + c);
    float* p = Agg + (long)d * F_DIM + c;
    atom_add_f32(p + 0, v.x);
    atom_add_f32(p + 1, v.y);
    atom_add_f32(p + 2, v.z);
    atom_add_f32(p + 3, v.w);
}

// ---------------------------------------------------------------------------
// 3) Out = relu(Agg @ Wrel + Hprev @ Wroot + b), fp32 WMMA 16x16x4.
//    Block = 256 threads = 8 wave32; block owns 16 rows (exact: N%16==0);
//    wave w owns cols [16w,16w+16).  A/H tiles staged via async global->LDS.
//    Out may alias Agg (block writes only its own rows, fully staged first).
// ---------------------------------------------------------------------------
__global__ __launch_bounds__(256)
void node_gemm_kernel(const float* __restrict__ Hprev, const float* __restrict__ Agg,
                      const float* __restrict__ Wrel,  const float* __restrict__ Wroot,
                      const float* __restrict__ bias,  float* __restrict__ Out) {
    // pad rows to 132 floats: lane l hits bank (4*l + k) % 64 -> conflict free
    __shared__ float sAgg[16][132];
    __shared__ float sH[16][132];

    int tid  = threadIdx.x;
    int row0 = blockIdx.x * 16;

    // async stage of the 16x128 Agg and Hprev tiles (16B per issue)
    for (int i = tid; i < 16 * 32; i += 256) {
        int r = i >> 5;
        int c = (i & 31) << 2;
        long g = (long)(row0 + r) * F_DIM + c;
        async_copy_b128((unsigned)(uintptr_t)&sAgg[r][c], Agg + g);
        async_copy_b128((unsigned)(uintptr_t)&sH[r][c],   Hprev + g);
    }
    wait_async0();
    __syncthreads();

    int wave = tid >> 5;
    int lane = tid & 31;
    int half = lane >> 4;   // 0: K={0,1}, 1: K={2,3} within the 16x4 A frag
    int l    = lane & 15;
    int col0 = wave << 4;

    v8f acc = {};
    for (int kk = 0; kk < F_DIM; kk += 4) {
        int k0 = kk + half * 2;
        v2f a, b;
        // agg @ Wrel
        a.x = sAgg[l][k0];
        a.y = sAgg[l][k0 + 1];
        b.x = Wrel[(long)k0 * F_DIM + col0 + l];
        b.y = Wrel[(long)(k0 + 1) * F_DIM + col0 + l];
        acc = __builtin_amdgcn_wmma_f32_16x16x4_f32(false, a, false, b,
                                                    (short)0, acc, false, false);
        // h @ Wroot
        a.x = sH[l][k0];
        a.y = sH[l][k0 + 1];
        b.x = Wroot[(long)k0 * F_DIM + col0 + l];
        b.y = Wroot[(long)(k0 + 1) * F_DIM + col0 + l];
        acc = __builtin_amdgcn_wmma_f32_16x16x4_f32(false, a, false, b,
                                                    (short)0, acc, false, false);
    }

    float bv = bias[col0 + l];
    #pragma unroll
    for (int r = 0; r < 8; ++r) {
        int row = row0 + r + half * 8;   // C/D layout: VGPR r -> rows r / r+8
        float v = acc[r] + bv;
        v = v > 0.0f ? v : 0.0f;
        Out[(long)row * F_DIM + col0 + l] = v;   // grid exact, no guard
    }
}

// ---------------------------------------------------------------------------
// 4) per-graph max / sum / counts.  H >= 0 (post-ReLU) so int-pattern
//    atomicMax on zero-initialized gmax is exact.
// ---------------------------------------------------------------------------
__global__ __launch_bounds__(256)
void pool_kernel(const float* __restrict__ H, const int* __restrict__ batch,
                 float* __restrict__ gmax, float* __restrict__ gsum,
                 int* __restrict__ counts) {
    int tid = blockIdx.x * 256 + threadIdx.x;
    if (tid >= N_NODES * 32) return;
    int i = tid >> 5;
    int c = (tid & 31) << 2;
    int g = batch[i];
    float4 v = *(const float4*)(H + (long)i * F_DIM + c);
    int base = g * F_DIM + c;
    atomicMax((int*)(gmax + base + 0), __float_as_int(v.x));
    atomicMax((int*)(gmax + base + 1), __float_as_int(v.y));
    atomicMax((int*)(gmax + base + 2), __float_as_int(v.z));
    atomicMax((int*)(gmax + base + 3), __float_as_int(v.w));
    atom_add_f32(gsum + base + 0, v.x);
    atom_add_f32(gsum + base + 1, v.y);
    atom_add_f32(gsum + base + 2, v.z);
    atom_add_f32(gsum + base + 3, v.w);
    if ((tid & 31) == 0) atomicAdd(counts + g, 1);
}

// ---------------------------------------------------------------------------
// 5) MLP head + log_softmax over graph axis (single block, LDS resident)
// ---------------------------------------------------------------------------
__global__ __launch_bounds__(256)
void head_kernel(const float* __restrict__ gmax, const float* __restrict__ gsum,
                 const int* __restrict__ counts,
                 const float* __restrict__ lw1, const float* __restrict__ lb1,
                 const float* __restrict__ lw2, const float* __restrict__ lb2,
                 const float* __restrict__ lw3, const float* __restrict__ lb3,
                 float* __restrict__ out) {
    __shared__ float z1[N_GRAPHS * 128];
    __shared__ float z2[N_GRAPHS * 64];
    __shared__ float z3[N_GRAPHS * N_LABELS];
    int tid = threadIdx.x;

    for (int idx = tid; idx < N_GRAPHS * 128; idx += 256) {
        int g = idx >> 7, j = idx & 127;
        float inv = 1.0f / fmaxf((float)counts[g], 1.0f);
        float a = lb1[j];
        for (int k = 0; k < 128; ++k)
            a += gmax[g * 128 + k] * lw1[k * 128 + j];
        for (int k = 0; k < 128; ++k)
            a += (gsum[g * 128 + k] * inv) * lw1[(128 + k) * 128 + j];
        z1[idx] = fmaxf(a, 0.0f);
    }
    __syncthreads();

    for (int idx = tid; idx < N_GRAPHS * 64; idx += 256) {
        int g = idx >> 6, j = idx & 63;
        float a = lb2[j];
        for (int k = 0; k < 128; ++k)
            a += z1[g * 128 + k] * lw2[k * 64 + j];
        z2[idx] = fmaxf(a, 0.0f);
    }
    __syncthreads();

    for (int idx = tid; idx < N_GRAPHS * N_LABELS; idx += 256) {
        int g = idx / N_LABELS, j = idx % N_LABELS;
        float a = lb3[j];
        for (int k = 0; k < 64; ++k)
            a += z2[g * 64 + k] * lw3[k * N_LABELS + j];
        z3[idx] = a;
    }
    __syncthreads();

    if (tid < N_LABELS) {
        float m = -1e30f;
        for (int g = 0; g < N_GRAPHS; ++g) m = fmaxf(m, z3[g * N_LABELS + tid]);
        float s = 0.0f;
        for (int g = 0; g < N_GRAPHS; ++g) s += expf(z3[g * N_LABELS + tid] - m);
        float lse = m + logf(s);
        for (int g = 0; g < N_GRAPHS; ++g)
            out[g * N_LABELS + tid] = z3[g * N_LABELS + tid] - lse;
    }
}

// ---------------------------------------------------------------------------
extern "C" void kernel_launch(void* const* d_in, const int* in_sizes, int n_in,
                              void* d_out, int out_size, void* d_ws, size_t ws_size,
                              hipStream_t stream) {
    const int*   x      = (const int*)d_in[0];
    const int*   ei     = (const int*)d_in[1];   // [2, N_EDGES]
    const int*   batch  = (const int*)d_in[2];
    const float* emb    = (const float*)d_in[3];
    const float* w1_rel = (const float*)d_in[4];
    const float* w1_root= (const float*)d_in[5];
    const float* b1     = (const float*)d_in[6];
    const float* w2_rel = (const float*)d_in[7];
    const float* w2_root= (const float*)d_in[8];
    const float* b2     = (const float*)d_in[9];
    const float* lw1    = (const float*)d_in[10];
    const float* lb1    = (const float*)d_in[11];
    const float* lw2    = (const float*)d_in[12];
    const float* lb2    = (const float*)d_in[13];
    const float* lw3    = (const float*)d_in[14];
    const float* lb3    = (const float*)d_in[15];

    const int* src = ei;
    const int* dst = ei + N_EDGES;

    const size_t NF = (size_t)N_NODES * F_DIM;
    float* A     = (float*)d_ws;           // h buffer / agg buffer (ping)
    float* B     = A + NF;                 // agg buffer / h buffer (pong)
    float* gmax  = B + NF;
    float* gsum  = gmax + N_GRAPHS * F_DIM;
    int*   cnt   = (int*)(gsum + N_GRAPHS * F_DIM);

    const int nodeThreads = N_NODES * 32;              // 3.2M
    const long edgeThreads = (long)N_EDGES * 32;       // 51.2M
    dim3 blk(256);
    dim3 gNode((nodeThreads + 255) / 256);
    dim3 gEdge((unsigned)((edgeThreads + 255) / 256));
    dim3 gGemm(N_NODES / 16);                          // 6250, exact

    // zero scratch: agg buffer B + pooling region
    hipMemsetAsync(B, 0, NF * sizeof(float), stream);
    hipMemsetAsync(gmax, 0, (size_t)(2 * N_GRAPHS * F_DIM) * sizeof(float)
                            + N_GRAPHS * sizeof(int), stream);

    // h0 = emb[x]
    gather_embed_kernel<<<gNode, blk, 0, stream>>>(x, emb, A);
    // layer 1: B = segment_sum(A[src] -> dst); B = relu(B@w1_rel + A@w1_root + b1)
    scatter_edges_kernel<<<gEdge, blk, 0, stream>>>(A, src, dst, B);
    node_gemm_kernel<<<gGemm, blk, 0, stream>>>(A, B, w1_rel, w1_root, b1, B);
    // layer 2: A = segment_sum(B[src] -> dst); A = relu(A@w2_rel + B@w2_root + b2)
    hipMemsetAsync(A, 0, NF * sizeof(float), stream);
    scatter_edges_kernel<<<gEdge, blk, 0, stream>>>(B, src, dst, A);
    node_gemm_kernel<<<gGemm, blk, 0, stream>>>(B, A, w2_rel, w2_root, b2, A);
    // pooling + head
    pool_kernel<<<gNode, blk, 0, stream>>>(A, batch, gmax, gsum, cnt);
    head_kernel<<<1, blk, 0, stream>>>(gmax, gsum, cnt, lw1, lb1, lw2, lb2,
                                       lw3, lb3, (float*)d_out);
}